// MBGCN_46136538693924
// MI455X (gfx1250) — compile-verified
//
#include <hip/hip_runtime.h>
#include <hip/hip_bf16.h>

// ---------------- problem constants (match reference) ----------------
constexpr int U_N  = 50000;
constexpr int I_N  = 25000;
constexpr int D_   = 64;
constexpr int L_   = 2;
constexpr int R_   = 2;
constexpr int E_UI = 500000;
constexpr int E_TR = 600000;
constexpr int E_II = 400000;
constexpr int B_   = 512;
constexpr int K_   = 100;
constexpr int DT   = D_ * (L_ + 1);      // 192
constexpr float EPS_     = 1e-8f;
constexpr float LAMB_    = 0.5f;
constexpr float L2_NORM_ = 1e-4f;

typedef __attribute__((ext_vector_type(2))) float v2f;
typedef __attribute__((ext_vector_type(8))) float v8f;

// ---------------- small prep kernels ----------------

// per-user mixing coefficients:
//   cmix[u,r] = softmax(mgnn)[r]*deg[u,r]/(deg@w + EPS) / (deg[u,r]+EPS)
//   cs2 [u,r] = 1/(deg[u,r]+EPS)
__global__ void prep_user_kernel(const float* __restrict__ mgnn,
                                 const int* __restrict__ ubd,
                                 float* __restrict__ cmix,
                                 float* __restrict__ cs2) {
  int u = blockIdx.x * blockDim.x + threadIdx.x;
  if (u >= U_N) return;
  float m0 = mgnn[0], m1 = mgnn[1];
  float mx = fmaxf(m0, m1);
  float e0 = expf(m0 - mx), e1 = expf(m1 - mx);
  float inv = 1.0f / (e0 + e1);
  float w0 = e0 * inv, w1 = e1 * inv;
  float d0 = (float)ubd[u * 2 + 0], d1 = (float)ubd[u * 2 + 1];
  float total = d0 * w0 + d1 * w1;
  float it = 1.0f / (total + EPS_);
  cmix[u * 2 + 0] = (d0 * w0 * it) / (d0 + EPS_);
  cmix[u * 2 + 1] = (d1 * w1 * it) / (d1 + EPS_);
  cs2[u * 2 + 0] = 1.0f / (d0 + EPS_);
  cs2[u * 2 + 1] = 1.0f / (d1 + EPS_);
}

__global__ void count_deg_kernel(const int* __restrict__ cols,
                                 float* __restrict__ deg, int nE) {
  int e = blockIdx.x * blockDim.x + threadIdx.x;
  if (e < nE) atomicAdd(&deg[cols[e]], 1.0f);
}

__global__ void invert_eps_kernel(float* __restrict__ v, int n) {
  int i = blockIdx.x * blockDim.x + threadIdx.x;
  if (i < n) v[i] = 1.0f / (v[i] + EPS_);
}

__global__ void invert_int_eps_kernel(const int* __restrict__ s,
                                      float* __restrict__ d, int n) {
  int i = blockIdx.x * blockDim.x + threadIdx.x;
  if (i < n) d[i] = 1.0f / ((float)s[i] + EPS_);
}

// copy a [rows,64] matrix into a 64-wide column slice of a stride-192 matrix
__global__ void copy_slice64_kernel(float* __restrict__ dst, int dstStride,
                                    const float* __restrict__ src, int rows) {
  int idx = blockIdx.x * blockDim.x + threadIdx.x;
  int r = idx >> 6, d = idx & 63;
  if (r < rows) dst[(long long)r * dstStride + d] = src[(long long)r * 64 + d];
}

// ---------------- fused scaled segment-sum (scatter) ----------------
// dst[dstIdx[e], :] += scale[dstIdx[e]*sStride + sOff] * src[srcIdx[e], :]
// one block per edge, blockDim.x == feature dim (64 or 192)
__global__ void seg_scatter_kernel(float* __restrict__ dst, int dstStride,
                                   const float* __restrict__ src, int srcStride,
                                   const int* __restrict__ dstIdx,
                                   const int* __restrict__ srcIdx,
                                   const float* __restrict__ scale,
                                   int sStride, int sOff) {
  int e = blockIdx.x;
  int d = threadIdx.x;
  int dr = dstIdx[e];
  int sr = srcIdx[e];
  float s = scale[(long long)dr * sStride + sOff];
  atomicAdd(&dst[(long long)dr * dstStride + d],
            s * src[(long long)sr * srcStride + d]);
}

// ---------------- fp32 WMMA GEMM: C[M,N] = A[M,K] @ B[K,N] ----------------
// One wave (32 threads) computes a 16 x (16*NTILES) strip of C with
// V_WMMA_F32_16X16X4_F32. The A fragment (one b64 per k-step) is loaded once
// and reused across all NTILES column tiles; consecutive WMMAs write
// different accumulators so there is no D->D dependency chain.
// Optional rowIdx gathers A rows (used for the batched-user projection).
// Fragment layouts per CDNA5 ISA 7.12.2 (wave32):
//   A 16x4 : lanes 0-15 hold M=lane, v0=K+0 v1=K+1 ; lanes 16-31: v0=K+2 v1=K+3
//   B 4x16 : v0 lanes0-15 = row K+0, v1 = row K+1 ; lanes16-31 rows K+2/K+3
//   C 16x16: VGPR v -> M = v + 8*(lane>>4), N = lane&15
template <int NTILES>
__global__ void wmma_gemm_f32_kernel(const float* __restrict__ A, int lda,
                                     const float* __restrict__ B, int ldb,
                                     float* __restrict__ C, int ldc,
                                     int M, int Kdim,
                                     const int* __restrict__ rowIdx) {
  const int lane = threadIdx.x & 31;
  const int m    = lane & 15;
  const int kh   = lane >> 4;           // half-wave selector (0/1)
  const int row0 = blockIdx.x * 16;

  int ar = row0 + m;
  if (ar > M - 1) ar = M - 1;           // clamp for partial last tile
  if (rowIdx) ar = rowIdx[ar];
  const float* __restrict__ Arow = A + (long long)ar * lda;
  const float* __restrict__ Bln  = B + m;   // lane's column within each tile

  v8f acc[NTILES];
#pragma unroll
  for (int t = 0; t < NTILES; ++t) acc[t] = (v8f){};

  for (int k = 0; k < Kdim; k += 4) {
    v2f a;
    a.x = Arow[k + 2 * kh + 0];
    a.y = Arow[k + 2 * kh + 1];
    const float* __restrict__ Brow0 = Bln + (long long)(k + 2 * kh + 0) * ldb;
    const float* __restrict__ Brow1 = Bln + (long long)(k + 2 * kh + 1) * ldb;
#pragma unroll
    for (int t = 0; t < NTILES; ++t) {
      v2f b;
      b.x = Brow0[t * 16];
      b.y = Brow1[t * 16];
      acc[t] = __builtin_amdgcn_wmma_f32_16x16x4_f32(
          /*neg_a=*/false, a, /*neg_b=*/false, b,
          /*c_mod=*/(short)0, acc[t], /*reuse_a=*/false, /*reuse_b=*/false);
    }
  }

#pragma unroll
  for (int t = 0; t < NTILES; ++t) {
#pragma unroll
    for (int v = 0; v < 8; ++v) {
      int r = row0 + v + 8 * kh;
      if (r < M) C[(long long)r * ldc + t * 16 + m] = acc[t][v];
    }
  }
}

// ---------------- scoring kernels (one wave per (b,k) pair) ----------------

__global__ void score1_l2_kernel(const float* __restrict__ Uf,
                                 const float* __restrict__ If,
                                 const int* __restrict__ users,
                                 const int* __restrict__ items,
                                 float* __restrict__ out) {
  int wave = (blockIdx.x * blockDim.x + threadIdx.x) >> 5;
  int lane = threadIdx.x & 31;
  if (wave >= B_ * K_) return;
  int b = wave / K_;
  int u = users[b];
  int it = items[wave];
  const float* ur = Uf + (long long)u * DT;
  const float* ir = If + (long long)it * DT;
  float dot = 0.0f, su = 0.0f, si = 0.0f;
  for (int j = lane; j < DT; j += 32) {
    float a = ur[j], c = ir[j];
    dot += a * c; su += a * a; si += c * c;
  }
#pragma unroll
  for (int off = 16; off > 0; off >>= 1) {
    dot += __shfl_down(dot, off);
    su  += __shfl_down(su, off);
    si  += __shfl_down(si, off);
  }
  if (lane == 0) {
    out[wave] = LAMB_ * dot;
    // summing ||bu||^2 once per (b,k) pair == *K factor in the reference
    atomicAdd(&out[B_ * K_], L2_NORM_ * (su + si));
  }
}

__global__ void score2_add_kernel(const float* __restrict__ upb,
                                  const float* __restrict__ iproj,
                                  const int* __restrict__ items,
                                  float* __restrict__ out) {
  int wave = (blockIdx.x * blockDim.x + threadIdx.x) >> 5;
  int lane = threadIdx.x & 31;
  if (wave >= B_ * K_) return;
  int b = wave / K_;
  int it = items[wave];
  const float* ur = upb + (long long)b * DT;
  const float* ir = iproj + (long long)it * DT;
  float dot = 0.0f;
  for (int j = lane; j < DT; j += 32) dot += ur[j] * ir[j];
#pragma unroll
  for (int off = 16; off > 0; off >>= 1) dot += __shfl_down(dot, off);
  if (lane == 0) out[wave] += ((1.0f - LAMB_) / (float)R_) * dot;
}

// ---------------- host launch ----------------

static inline int ceil_div(int a, int b) { return (a + b - 1) / b; }

extern "C" void kernel_launch(void* const* d_in, const int* in_sizes, int n_in,
                              void* d_out, int out_size, void* d_ws, size_t ws_size,
                              hipStream_t stream) {
  (void)in_sizes; (void)n_in; (void)out_size; (void)ws_size;
  const float* ue    = (const float*)d_in[0];
  const float* ie    = (const float*)d_in[1];
  const float* Wui   = (const float*)d_in[2];    // [L,64,64]
  const float* Wii   = (const float*)d_in[3];    // [R,L,64,64]
  const float* Wb    = (const float*)d_in[4];    // [R,192,192]
  const float* mgnn  = (const float*)d_in[5];    // [R]
  const int* rel_rows = (const int*)d_in[6];     // [R,E_UI]
  const int* rel_cols = (const int*)d_in[7];
  const int* tr_rows  = (const int*)d_in[8];     // [E_TR]
  const int* tr_cols  = (const int*)d_in[9];
  const int* ig_rows  = (const int*)d_in[10];    // [R,E_II]
  const int* ig_cols  = (const int*)d_in[11];
  const int* ubd      = (const int*)d_in[12];    // [U,R]
  const int* igd      = (const int*)d_in[13];    // [R,I]
  const int* users    = (const int*)d_in[14];    // [B]
  const int* items    = (const int*)d_in[15];    // [B,K]
  float* out = (float*)d_out;                    // [B*K + 1]

  // ---- workspace carve-up (floats) ----
  float* ws = (float*)d_ws;
  size_t o = 0;
  float* Uf    = ws + o; o += (size_t)U_N * DT;   // user concat  [U,192]
  float* If    = ws + o; o += (size_t)I_N * DT;   // item concat  [I,192]
  float* S0    = ws + o; o += (size_t)I_N * DT;   // S_final[0]
  float* S1    = ws + o; o += (size_t)I_N * DT;   // S_final[1]
  float* Unb   = ws + o; o += (size_t)U_N * D_;   // user neighbor agg
  float* Ius   = ws + o; o += (size_t)I_N * D_;   // item<-user agg
  float* Stmp  = ws + o; o += (size_t)I_N * D_;   // item-item agg
  float* agg2  = ws + o; o += (size_t)U_N * DT;   // score2 user agg
  float* iproj = ws + o; o += (size_t)I_N * DT;   // item_proj
  float* upb   = ws + o; o += (size_t)B_ * DT;    // user_proj[users]
  float* cmix  = ws + o; o += (size_t)U_N * R_;   // mixing coefs
  float* cs2   = ws + o; o += (size_t)U_N * R_;   // 1/(ubd+eps)
  float* invd  = ws + o; o += (size_t)I_N;        // 1/(item_deg+eps)
  float* invg  = ws + o; o += (size_t)R_ * I_N;   // 1/(igd+eps)

  float* Sf[2] = {S0, S1};

  // ---- prep ----
  prep_user_kernel<<<ceil_div(U_N, 256), 256, 0, stream>>>(mgnn, ubd, cmix, cs2);
  hipMemsetAsync(invd, 0, (size_t)I_N * sizeof(float), stream);
  count_deg_kernel<<<ceil_div(E_TR, 256), 256, 0, stream>>>(tr_cols, invd, E_TR);
  invert_eps_kernel<<<ceil_div(I_N, 256), 256, 0, stream>>>(invd, I_N);
  invert_int_eps_kernel<<<ceil_div(R_ * I_N, 256), 256, 0, stream>>>(igd, invg, R_ * I_N);

  copy_slice64_kernel<<<ceil_div(U_N * 64, 256), 256, 0, stream>>>(Uf, DT, ue, U_N);
  copy_slice64_kernel<<<ceil_div(I_N * 64, 256), 256, 0, stream>>>(If, DT, ie, I_N);
  copy_slice64_kernel<<<ceil_div(I_N * 64, 256), 256, 0, stream>>>(S0, DT, ie, I_N);
  copy_slice64_kernel<<<ceil_div(I_N * 64, 256), 256, 0, stream>>>(S1, DT, ie, I_N);

  const int uTiles = ceil_div(U_N, 16);   // 3125
  const int iTiles = ceil_div(I_N, 16);   // 1563
  const int bTiles = ceil_div(B_, 16);    // 32

  // ---- propagation (L layers) ----
  for (int l = 0; l < L_; ++l) {
    const float* Ucur = Uf + 64 * l;   // stride DT
    const float* Icur = If + 64 * l;   // stride DT

    // user <- item per behavior, scale folded in
    hipMemsetAsync(Unb, 0, (size_t)U_N * D_ * sizeof(float), stream);
    for (int r = 0; r < R_; ++r) {
      seg_scatter_kernel<<<E_UI, D_, 0, stream>>>(
          Unb, D_, Icur, DT,
          rel_rows + (size_t)r * E_UI, rel_cols + (size_t)r * E_UI,
          cmix, R_, r);
    }
    wmma_gemm_f32_kernel<4><<<uTiles, 32, 0, stream>>>(
        Unb, D_, Wui + (size_t)l * D_ * D_, D_,
        Uf + 64 * (l + 1), DT, U_N, D_, nullptr);

    // item <- user through train matrix transpose, 1/item_deg folded in
    hipMemsetAsync(Ius, 0, (size_t)I_N * D_ * sizeof(float), stream);
    seg_scatter_kernel<<<E_TR, D_, 0, stream>>>(
        Ius, D_, Ucur, DT, tr_cols, tr_rows, invd, 1, 0);
    wmma_gemm_f32_kernel<4><<<iTiles, 32, 0, stream>>>(
        Ius, D_, Wui + (size_t)l * D_ * D_, D_,
        If + 64 * (l + 1), DT, I_N, D_, nullptr);

    // item-item propagation per behavior
    for (int r = 0; r < R_; ++r) {
      hipMemsetAsync(Stmp, 0, (size_t)I_N * D_ * sizeof(float), stream);
      seg_scatter_kernel<<<E_II, D_, 0, stream>>>(
          Stmp, D_, Icur, DT,
          ig_rows + (size_t)r * E_II, ig_cols + (size_t)r * E_II,
          invg + (size_t)r * I_N, 1, 0);
      wmma_gemm_f32_kernel<4><<<iTiles, 32, 0, stream>>>(
          Stmp, D_, Wii + ((size_t)r * L_ + l) * D_ * D_, D_,
          Sf[r] + 64 * (l + 1), DT, I_N, D_, nullptr);
    }
  }

  // ---- scoring ----
  hipMemsetAsync(out, 0, (size_t)(B_ * K_ + 1) * sizeof(float), stream);
  score1_l2_kernel<<<ceil_div(B_ * K_ * 32, 256), 256, 0, stream>>>(
      Uf, If, users, items, out);

  for (int r = 0; r < R_; ++r) {
    hipMemsetAsync(agg2, 0, (size_t)U_N * DT * sizeof(float), stream);
    seg_scatter_kernel<<<E_UI, DT, 0, stream>>>(
        agg2, DT, Sf[r], DT,
        rel_rows + (size_t)r * E_UI, rel_cols + (size_t)r * E_UI,
        cs2, R_, r);
    // user_proj[users]  = agg2[users] @ Wb[r]   (row gather inside GEMM)
    wmma_gemm_f32_kernel<12><<<bTiles, 32, 0, stream>>>(
        agg2, DT, Wb + (size_t)r * DT * DT, DT, upb, DT, B_, DT, users);
    // item_proj = Sf[r] @ Wb[r]
    wmma_gemm_f32_kernel<12><<<iTiles, 32, 0, stream>>>(
        Sf[r], DT, Wb + (size_t)r * DT * DT, DT, iproj, DT, I_N, DT, nullptr);
    score2_add_kernel<<<ceil_div(B_ * K_ * 32, 256), 256, 0, stream>>>(
        upb, iproj, items, out);
  }
}